// FirstOrderCondRNN_18932215841504
// MI455X (gfx1250) — compile-verified
//
#include <hip/hip_runtime.h>

#define TPB   256
#define N_KC  200
#define N_MBON 20
#define N_FBN  60
#define N_DAN  20
#define NR    100
#define TT     61
#define NW   (N_MBON * N_KC)   // 4000
#define NV   (NW / 4)          // 1000 vec4s
#define WR_LD 101              // padded Wr row stride (conflict-free in LDS)

// LDS partition (floats)
#define OFF_RKC   0                       // [TT*N_KC] transposed: [t*200 + k]
#define OFF_WR    (OFF_RKC + TT * N_KC)   // [100*101]
#define OFF_P     (OFF_WR + NR * WR_LD)   // [1000]
#define OFF_Q     (OFF_P + NV)            // [200]
#define OFF_R     (OFF_Q + N_KC)          // [112]
#define OFF_WRP   (OFF_R + 112)           // [112]
#define OFF_RBK   (OFF_WRP + 112)         // [200]
#define OFF_RBD   (OFF_RBK + N_KC)        // [32]
#define OFF_IFBN  (OFF_RBD + 32)          // [64]
#define OFF_BIAS  (OFF_IFBN + 64)         // [112]
#define OFF_WRO   (OFF_BIAS + 112)        // [32]
#define OFF_WEXT  (OFF_WRO + 32)          // [128]
#define SMEM_FLOATS (OFF_WEXT + 128)      // 24292 floats = 97168 B

typedef float v2f __attribute__((ext_vector_type(2)));
typedef float v4f __attribute__((ext_vector_type(4)));
typedef float v8f __attribute__((ext_vector_type(8)));

__global__ __launch_bounds__(TPB) void cond_rnn_kernel(
    const float* __restrict__ r_kc, const float* __restrict__ r_ext,
    const float* __restrict__ timev,
    const float* __restrict__ W0_W, const float* __restrict__ W0_w,
    const float* __restrict__ W_recur, const float* __restrict__ W_ext,
    const float* __restrict__ bias, const float* __restrict__ W_readout,
    float* __restrict__ o_r, float* __restrict__ o_W,
    float* __restrict__ o_wt, float* __restrict__ o_ro, int Bn)
{
  extern __shared__ float smem[];
  float* s_rkc  = smem + OFF_RKC;
  float* s_wr   = smem + OFF_WR;
  float* s_P    = smem + OFF_P;
  float* s_Q    = smem + OFF_Q;
  float* s_r    = smem + OFF_R;
  float* s_wrp  = smem + OFF_WRP;
  float* s_rbk  = smem + OFF_RBK;
  float* s_rbd  = smem + OFF_RBD;
  float* s_ifbn = smem + OFF_IFBN;
  float* s_bias = smem + OFF_BIAS;
  float* s_wro  = smem + OFF_WRO;
  float* s_wext = smem + OFF_WEXT;

  const int tid  = threadIdx.x;
  const int b    = blockIdx.x;
  const int lane = tid & 31;
  const int wid  = tid >> 5;

  const float dt = timev[1] - timev[0];
  const float cr = dt / 1.0f;   // dt / TAU_R
  const float cw = dt / 5.0f;   // dt / TAU_W

  // ---- Preload r_kc slice, transposed to [t][k] ----
  const float* g_rkc = r_kc + (size_t)b * (N_KC * TT);
  for (int gi = tid; gi < (N_KC * TT) / 4; gi += TPB) {  // 3050 vec4s
    v4f v = *(const v4f*)(g_rkc + 4 * gi);
    int f = 4 * gi;
    #pragma unroll
    for (int e = 0; e < 4; ++e) {
      int ff = f + e;
      int k = ff / TT, t = ff % TT;
      s_rkc[t * N_KC + k] = v[e];
    }
  }
  // ---- Preload Wr (zero MBON<-DAN block), padded stride 101 ----
  for (int gi = tid; gi < (NR * NR) / 4; gi += TPB) {    // 2500 vec4s
    v4f v = *(const v4f*)(W_recur + 4 * gi);
    int s = gi / 25;
    int r0 = (gi % 25) * 4;
    #pragma unroll
    for (int e = 0; e < 4; ++e) {
      int r = r0 + e;
      float val = v[e];
      if (s < N_MBON && r >= NR - N_DAN) val = 0.0f;
      s_wr[s * WR_LD + r] = val;
    }
  }
  if (tid < NR)        s_bias[tid] = bias[tid];
  if (tid < N_MBON)    s_wro[tid]  = W_readout[tid];
  if (tid < N_FBN * 2) s_wext[tid] = W_ext[tid];

  // ---- Plastic weights resident in registers (vec4 layout) ----
  v4f W4[4], wt4[4];
  const float* gW0 = W0_W + (size_t)b * NW;
  const float* gw0 = W0_w + (size_t)b * NW;
  #pragma unroll
  for (int p = 0; p < 4; ++p) {
    int v = p * TPB + tid;
    if (v < NV) {
      W4[p]  = *(const v4f*)(gW0 + 4 * v);
      wt4[p] = *(const v4f*)(gw0 + 4 * v);
    } else {
      W4[p]  = (v4f){0.f, 0.f, 0.f, 0.f};
      wt4[p] = (v4f){0.f, 0.f, 0.f, 0.f};
    }
  }
  __syncthreads();

  // ---- Init state ----
  if (tid < 112)  s_r[tid]   = (tid < N_MBON) ? 0.0f : ((tid < NR) ? 0.1f : 0.0f);
  if (tid < N_DAN) s_rbd[tid] = 0.1f;
  if (tid < N_KC)  s_rbk[tid] = s_rkc[tid];   // r_kc[:, :, 0]

  // ---- Hoist constant WMMA A-fragments of Wr into registers ----
  // A 16x4 f32 layout: lanes 0-15 hold K={0,1} (vgpr0,1); lanes 16-31 K={2,3}
  v2f afrag[25];
  {
    int row  = 16 * wid + (lane & 15);
    int ko   = (lane >> 4) * 2;
    bool valid = (wid < 7) && (row < NR);
    int rowc = valid ? row : 0;
    #pragma unroll
    for (int j = 0; j < 25; ++j) {
      float a0 = s_wr[rowc * WR_LD + 4 * j + ko];
      float a1 = s_wr[rowc * WR_LD + 4 * j + ko + 1];
      afrag[j][0] = valid ? a0 : 0.0f;
      afrag[j][1] = valid ? a1 : 0.0f;
    }
  }

  // ---- t = 0 series stores ----
  if (tid < NR) o_r[(size_t)b * NR + tid] = s_r[tid];
  if (tid == 0) o_ro[b] = 0.0f;  // W_readout @ zeros
  {
    float* oW  = o_W  + (size_t)b * NW;
    float* owt = o_wt + (size_t)b * NW;
    #pragma unroll
    for (int p = 0; p < 4; ++p) {
      int v = p * TPB + tid;
      if (v < NV) {
        __builtin_nontemporal_store(W4[p],  (v4f*)(oW  + 4 * v));
        __builtin_nontemporal_store(wt4[p], (v4f*)(owt + 4 * v));
      }
    }
  }
  __syncthreads();

  // ==== time loop ====
  for (int t = 0; t < TT - 1; ++t) {
    const float* rkt = s_rkc + t * N_KC;
    v4f rk4[4];

    // ---- phase 1: I_kc partials, rbk EMA, I_fbn, WMMA Wr@r ----
    #pragma unroll
    for (int p = 0; p < 4; ++p) {
      int v = p * TPB + tid;
      if (v < NV) {
        int k = (v % 50) * 4;
        rk4[p] = *(const v4f*)(rkt + k);
        v4f w = W4[p];
        s_P[v] = w[0] * rk4[p][0] + w[1] * rk4[p][1] +
                 w[2] * rk4[p][2] + w[3] * rk4[p][3];
      }
    }
    if (tid < N_KC) {
      float rb = s_rbk[tid];
      s_rbk[tid] = rb + (rkt[tid] - rb) * cw;
    }
    if (tid < N_FBN) {
      float re0 = r_ext[(size_t)b * (2 * TT) + t];
      float re1 = r_ext[(size_t)b * (2 * TT) + TT + t];
      s_ifbn[tid] = s_wext[2 * tid] * re0 + s_wext[2 * tid + 1] * re1;
    }
    if (wid < 7) {
      v8f acc = {0.f, 0.f, 0.f, 0.f, 0.f, 0.f, 0.f, 0.f};
      bool first = (lane & 15) == 0;
      #pragma unroll
      for (int j = 0; j < 25; ++j) {
        // B 4x16: vgpr0 = rows K={0,2} (lane halves), vgpr1 = rows K={1,3};
        // only column N=0 carries r.
        int ra = 4 * j + ((lane >> 4) << 1);
        v2f rv = *(const v2f*)(s_r + ra);
        v2f bfrag;
        bfrag[0] = first ? rv[0] : 0.0f;
        bfrag[1] = first ? rv[1] : 0.0f;
        acc = __builtin_amdgcn_wmma_f32_16x16x4_f32(
            false, afrag[j], false, bfrag, (short)0, acc, false, false);
      }
      if (first) {  // lane0: M=0..7, lane16: M=8..15 (column N=0)
        int base = 16 * wid + (lane >> 4) * 8;
        v4f lo = {acc[0], acc[1], acc[2], acc[3]};
        v4f hi = {acc[4], acc[5], acc[6], acc[7]};
        *(v4f*)(s_wrp + base)     = lo;
        *(v4f*)(s_wrp + base + 4) = hi;
      }
    }
    __syncthreads();

    // ---- phase 2: reduce partials (1000 -> 200) ----
    if (tid < N_KC) {
      int d = tid / 10, j0 = tid % 10;
      const float* pp = s_P + 50 * d + 5 * j0;
      s_Q[tid] = pp[0] + pp[1] + pp[2] + pp[3] + pp[4];
    }
    __syncthreads();

    // ---- phase 3: r update + r_series store ----
    if (tid < NR) {
      float it;
      if (tid < N_MBON) {
        const float* q = s_Q + 10 * tid;
        it = q[0] + q[1] + q[2] + q[3] + q[4] + q[5] + q[6] + q[7] + q[8] + q[9];
      } else if (tid < N_MBON + N_FBN) {
        it = s_ifbn[tid - N_MBON];
      } else {
        it = 0.0f;
      }
      float pre  = s_wrp[tid] + s_bias[tid] + it;
      float relu = pre > 0.0f ? pre : 0.0f;
      float ro   = s_r[tid];
      float rn   = ro + (relu - ro) * cr;
      s_r[tid] = rn;
      o_r[((size_t)(t + 1) * Bn + b) * NR + tid] = rn;
    }
    __syncthreads();

    // ---- phase 4: rbd EMA + readout ----
    if (tid < N_DAN) {
      float rb = s_rbd[tid];
      s_rbd[tid] = rb + (s_r[NR - N_DAN + tid] - rb) * cw;
    }
    if (tid == 0) {
      float acc = 0.0f;
      #pragma unroll
      for (int m = 0; m < N_MBON; ++m) acc += s_wro[m] * s_r[m];
      o_ro[(size_t)(t + 1) * Bn + b] = acc;
    }
    __syncthreads();

    // ---- phase 5: plasticity update + streaming series stores ----
    {
      float* oW  = o_W  + ((size_t)(t + 1) * Bn + b) * NW;
      float* owt = o_wt + ((size_t)(t + 1) * Bn + b) * NW;
      #pragma unroll
      for (int p = 0; p < 4; ++p) {
        int v = p * TPB + tid;
        if (v < NV) {
          int d = v / 50;
          int k = (v % 50) * 4;
          v4f rbk4 = *(const v4f*)(s_rbk + k);
          float sd = s_rbd[d];
          float rd = s_r[NR - N_DAN + d];
          v4f w = W4[p], wt = wt4[p], rk = rk4[p];
          #pragma unroll
          for (int e = 0; e < 4; ++e) {
            wt[e] += (sd * rk[e] - rd * rbk4[e]) * dt;
            w[e] = fminf(fmaxf(w[e] + (wt[e] - w[e]) * cw, 0.0f), 0.05f);
          }
          wt4[p] = wt;
          W4[p]  = w;
          __builtin_nontemporal_store(wt, (v4f*)(owt + 4 * v));
          __builtin_nontemporal_store(w,  (v4f*)(oW  + 4 * v));
        }
      }
    }
    __syncthreads();
  }
}

extern "C" void kernel_launch(void* const* d_in, const int* in_sizes, int n_in,
                              void* d_out, int out_size, void* d_ws, size_t ws_size,
                              hipStream_t stream) {
  (void)n_in; (void)out_size; (void)d_ws; (void)ws_size;
  const float* r_kc      = (const float*)d_in[0];
  const float* r_ext     = (const float*)d_in[1];
  const float* timev     = (const float*)d_in[2];
  // d_in[3] = n_batch scalar (host derives B from sizes instead)
  const float* W0_W      = (const float*)d_in[4];
  const float* W0_w      = (const float*)d_in[5];
  const float* W_recur   = (const float*)d_in[6];
  const float* W_ext     = (const float*)d_in[7];
  const float* bias      = (const float*)d_in[8];
  const float* W_readout = (const float*)d_in[9];

  int Bn = in_sizes[0] / (N_KC * TT);

  float* out  = (float*)d_out;
  float* o_r  = out;
  float* o_W  = o_r  + (size_t)TT * Bn * NR;
  float* o_wt = o_W  + (size_t)TT * Bn * NW;
  float* o_ro = o_wt + (size_t)TT * Bn * NW;

  size_t shmem = (size_t)SMEM_FLOATS * sizeof(float);
  cond_rnn_kernel<<<dim3(Bn), dim3(TPB), shmem, stream>>>(
      r_kc, r_ext, timev, W0_W, W0_w, W_recur, W_ext, bias, W_readout,
      o_r, o_W, o_wt, o_ro, Bn);
}